// SCGNNModel_3642132267298
// MI455X (gfx1250) — compile-verified
//
#include <hip/hip_runtime.h>

// ---------------------------------------------------------------------------
// 2-layer GCN on MI455X (gfx1250, wave32).
//   h1 = relu( Dinv (A+I) Dinv (X W1) + b1 )
//   y  =       Dinv (A+I) Dinv (h1 W2) + b2
// Factored as: p = dinv[s] * (X W)[s];  agg[d] = sum_edges p[src];
//              out[d] = dinv[d]*(agg[d] + p[d]) + b
// Dense matmuls use V_WMMA_F32_16X16X4_F32 (exact fp32). Scatter phase uses
// GLOBAL_ATOMIC_ADD_F32; whole working set (<52MB) lives in the 192MB L2.
// ---------------------------------------------------------------------------

#define SENSORS 50000
#define X_DIM   64
#define HID     64
#define Y_DIM   32
#define N_EDGES 800000
#define NP      50048   // SENSORS padded to multiple of 64 (buffer alignment)

typedef float v2f __attribute__((ext_vector_type(2)));
typedef float v8f __attribute__((ext_vector_type(8)));

// ---- init: deg=1 (self loop), zero aggregation buffers ---------------------
__global__ __launch_bounds__(256) void scgnn_init(float* __restrict__ deg,
                                                  float* __restrict__ agg1,
                                                  float* __restrict__ agg2,
                                                  int n) {
    int idx = blockIdx.x * 256 + threadIdx.x;
    if (idx < n * HID)   agg1[idx] = 0.0f;
    if (idx < n * Y_DIM) agg2[idx] = 0.0f;
    if (idx < n)         deg[idx]  = 1.0f;   // self loop contributes 1
}

// ---- degree count over dst -------------------------------------------------
__global__ __launch_bounds__(256) void scgnn_deg(const int* __restrict__ ei,
                                                 float* __restrict__ deg, int E) {
    int e = blockIdx.x * 256 + threadIdx.x;
    if (e < E) atomicAdd(&deg[ei[E + e]], 1.0f);
}

// ---- dinv = rsqrt(deg) in place (deg >= 1 always) --------------------------
__global__ __launch_bounds__(256) void scgnn_rsqrt(float* __restrict__ deg, int n) {
    int i = blockIdx.x * 256 + threadIdx.x;
    if (i < n) deg[i] = rsqrtf(deg[i]);
}

// ---- GEMM1: P1[s][0:64] = dinv[s] * sum_k X[s][k] W1[k][0:64] --------------
// X is column-major: X[s][k] = x[k*N + s]  (x_input is [64, SENSORS])
// One wave = one 16x16 output tile; 8 waves/block = 32 rows x 64 cols.
__global__ __launch_bounds__(256) void scgnn_gemm1(const float* __restrict__ x,
                                                   const float* __restrict__ W,
                                                   const float* __restrict__ dinv,
                                                   float* __restrict__ P, int N) {
    const int lane = threadIdx.x & 31;
    const int wave = threadIdx.x >> 5;
    const int n0   = (wave & 3) * 16;                    // col tile 0..3
    const int m0   = (blockIdx.x * 2 + (wave >> 2)) * 16; // row tile
    if (m0 >= N) return;                                  // wave-uniform exit

    const int lh = lane >> 4;       // 0: K pair {0,1} / 1: K pair {2,3}
    const int ll = lane & 15;
    int mrow = m0 + ll; if (mrow >= N) mrow = N - 1;      // clamp (masked later)

    v8f c = {0.f, 0.f, 0.f, 0.f, 0.f, 0.f, 0.f, 0.f};
    #pragma unroll
    for (int k = 0; k < X_DIM; k += 4) {
        const int ka = k + 2 * lh;
        v2f a, b;
        a.x = x[(size_t)ka * N + mrow];          // A[m][ka]
        a.y = x[(size_t)(ka + 1) * N + mrow];    // A[m][ka+1]
        b.x = W[ka * HID + n0 + ll];             // B[ka][n]
        b.y = W[(ka + 1) * HID + n0 + ll];       // B[ka+1][n]
        c = __builtin_amdgcn_wmma_f32_16x16x4_f32(false, a, false, b,
                                                  (short)0, c, false, false);
    }
    #pragma unroll
    for (int r = 0; r < 8; ++r) {                // C/D: VGPR r -> M = r + 8*lh
        const int m = m0 + r + 8 * lh;
        if (m < N) P[(size_t)m * HID + n0 + ll] = dinv[m] * c[r];
    }
}

// ---- edge scatter, 64-wide: agg1[dst] += p1[src] (16 threads/edge, float4) -
__global__ __launch_bounds__(256) void scgnn_scatter64(const int* __restrict__ ei,
                                                       const float* __restrict__ p,
                                                       float* __restrict__ agg, int E) {
    long long t = (long long)blockIdx.x * 256 + threadIdx.x;
    int e  = (int)(t >> 4);
    int f4 = ((int)t & 15) << 2;
    if (e < E) {
        const int src = ei[e], dst = ei[E + e];
        const float4 v = *(const float4*)(p + (size_t)src * 64 + f4);
        float* o = agg + (size_t)dst * 64 + f4;
        atomicAdd(o + 0, v.x); atomicAdd(o + 1, v.y);
        atomicAdd(o + 2, v.z); atomicAdd(o + 3, v.w);
    }
}

// ---- epilogue1: act = relu(dinv*(agg1 + p1) + b1), written over agg1 -------
__global__ __launch_bounds__(256) void scgnn_epi1(const float* __restrict__ p1,
                                                  float* __restrict__ agg1,
                                                  const float* __restrict__ dinv,
                                                  const float* __restrict__ b1, int n) {
    int idx = blockIdx.x * 256 + threadIdx.x;
    if (idx < n * HID) {
        const int s = idx >> 6, f = idx & 63;
        float v = dinv[s] * (agg1[idx] + p1[idx]) + b1[f];
        agg1[idx] = v > 0.0f ? v : 0.0f;
    }
}

// ---- GEMM2: P2[s][0:32] = dinv[s] * sum_k act[s][k] W2[k][0:32] ------------
// act is row-major [N][64]; 8 waves/block = 64 rows x 32 cols.
__global__ __launch_bounds__(256) void scgnn_gemm2(const float* __restrict__ act,
                                                   const float* __restrict__ W,
                                                   const float* __restrict__ dinv,
                                                   float* __restrict__ P, int N) {
    const int lane = threadIdx.x & 31;
    const int wave = threadIdx.x >> 5;
    const int n0   = (wave & 1) * 16;                     // col tile 0..1
    const int m0   = (blockIdx.x * 4 + (wave >> 1)) * 16; // row tile
    if (m0 >= N) return;                                  // wave-uniform exit

    const int lh = lane >> 4;
    const int ll = lane & 15;
    int mrow = m0 + ll; if (mrow >= N) mrow = N - 1;

    v8f c = {0.f, 0.f, 0.f, 0.f, 0.f, 0.f, 0.f, 0.f};
    #pragma unroll
    for (int k = 0; k < HID; k += 4) {
        const int ka = k + 2 * lh;
        v2f a, b;
        a.x = act[(size_t)mrow * HID + ka];
        a.y = act[(size_t)mrow * HID + ka + 1];
        b.x = W[ka * Y_DIM + n0 + ll];
        b.y = W[(ka + 1) * Y_DIM + n0 + ll];
        c = __builtin_amdgcn_wmma_f32_16x16x4_f32(false, a, false, b,
                                                  (short)0, c, false, false);
    }
    #pragma unroll
    for (int r = 0; r < 8; ++r) {
        const int m = m0 + r + 8 * lh;
        if (m < N) P[(size_t)m * Y_DIM + n0 + ll] = dinv[m] * c[r];
    }
}

// ---- edge scatter, 32-wide: agg2[dst] += p2[src] (8 threads/edge, float4) --
__global__ __launch_bounds__(256) void scgnn_scatter32(const int* __restrict__ ei,
                                                       const float* __restrict__ p,
                                                       float* __restrict__ agg, int E) {
    long long t = (long long)blockIdx.x * 256 + threadIdx.x;
    int e  = (int)(t >> 3);
    int f4 = ((int)t & 7) << 2;
    if (e < E) {
        const int src = ei[e], dst = ei[E + e];
        const float4 v = *(const float4*)(p + (size_t)src * 32 + f4);
        float* o = agg + (size_t)dst * 32 + f4;
        atomicAdd(o + 0, v.x); atomicAdd(o + 1, v.y);
        atomicAdd(o + 2, v.z); atomicAdd(o + 3, v.w);
    }
}

// ---- epilogue2: out[y][s] = dinv[s]*(agg2[s][y] + p2[s][y]) + b2[y] --------
// gridDim.y = Y_DIM -> coalesced writes of the transposed output.
__global__ __launch_bounds__(256) void scgnn_epi2(const float* __restrict__ p2,
                                                  const float* __restrict__ agg2,
                                                  const float* __restrict__ dinv,
                                                  const float* __restrict__ b2,
                                                  float* __restrict__ out, int n) {
    int s = blockIdx.x * 256 + threadIdx.x;
    int y = blockIdx.y;
    if (s < n) {
        size_t i = (size_t)s * Y_DIM + y;
        out[(size_t)y * n + s] = dinv[s] * (agg2[i] + p2[i]) + b2[y];
    }
}

extern "C" void kernel_launch(void* const* d_in, const int* in_sizes, int n_in,
                              void* d_out, int out_size, void* d_ws, size_t ws_size,
                              hipStream_t stream) {
    const float* x  = (const float*)d_in[0];   // [1, 64, SENSORS]
    const float* W1 = (const float*)d_in[1];   // [64, 64]
    const float* b1 = (const float*)d_in[2];   // [64]
    const float* W2 = (const float*)d_in[3];   // [64, 32]
    const float* b2 = (const float*)d_in[4];   // [32]
    const int*   ei = (const int*)d_in[5];     // [2, N_EDGES]
    float* out = (float*)d_out;                // [1, 32, SENSORS]

    const int N = SENSORS, E = N_EDGES;

    // Workspace carve-up (floats): dinv | p1 | agg1/act | p2 | agg2
    float* ws   = (float*)d_ws;
    float* dinv = ws;                       // NP
    float* p1   = ws + (size_t)NP;          // NP*64
    float* agg1 = ws + (size_t)NP * 65;     // NP*64 (reused as act after epi1)
    float* p2   = ws + (size_t)NP * 129;    // NP*32
    float* agg2 = ws + (size_t)NP * 161;    // NP*32   (total NP*193*4 ~ 38.6MB)

    scgnn_init<<<(N * HID + 255) / 256, 256, 0, stream>>>(dinv, agg1, agg2, N);
    scgnn_deg<<<(E + 255) / 256, 256, 0, stream>>>(ei, dinv, E);
    scgnn_rsqrt<<<(N + 255) / 256, 256, 0, stream>>>(dinv, N);

    const int rowTiles = (N + 15) / 16;                    // 3125
    scgnn_gemm1<<<(rowTiles + 1) / 2, 256, 0, stream>>>(x, W1, dinv, p1, N);
    scgnn_scatter64<<<(int)(((long long)E * 16 + 255) / 256), 256, 0, stream>>>(ei, p1, agg1, E);
    scgnn_epi1<<<(N * HID + 255) / 256, 256, 0, stream>>>(p1, agg1, dinv, b1, N);

    scgnn_gemm2<<<(rowTiles + 3) / 4, 256, 0, stream>>>(agg1, W2, dinv, p2, N);
    scgnn_scatter32<<<(int)(((long long)E * 8 + 255) / 256), 256, 0, stream>>>(ei, p2, agg2, E);
    scgnn_epi2<<<dim3((N + 255) / 256, Y_DIM), 256, 0, stream>>>(p2, agg2, dinv, b2, out, N);
}